// FLAME_2808908612143
// MI455X (gfx1250) — compile-verified
//
#include <hip/hip_runtime.h>

// ---------------- constants ----------------
#define NB     2048          // batch
#define NV     5023          // vertices
#define NC     (NV*3)        // 15069 flattened columns
#define NSH    100
#define NEXP   50
#define KDIM   192           // 150 betas + 2 pad + 36 pose-feature + 4 pad
#define NJ     5
#define VTILES 314           // ceil(5023/16)

typedef __attribute__((ext_vector_type(2))) float v2f;
typedef __attribute__((ext_vector_type(8))) float v8f;

// ---------------------------------------------------------------
// Stage 1: pack B matrix [NC][KDIM]:
//   k<150         : shapedirs  (shapedirs[v,3,150] flattens to [NC,150] row-major)
//   150..151      : 0
//   152..187      : posedirs[k-152][c]   (transpose gather)
//   188..191      : 0
// ---------------------------------------------------------------
__global__ void build_B(const float* __restrict__ sd, const float* __restrict__ pd,
                        float* __restrict__ Bfull) {
    int idx = blockIdx.x * blockDim.x + threadIdx.x;
    const int total = NC * KDIM;
    const int stride = gridDim.x * blockDim.x;
    for (; idx < total; idx += stride) {
        int c = idx / KDIM, k = idx % KDIM;
        float v;
        if (k < 150)                 v = sd[c * 150 + k];
        else if (k >= 152 && k < 188) v = pd[(k - 152) * NC + c];
        else                          v = 0.0f;
        Bfull[idx] = v;
    }
}

// ---------------------------------------------------------------
// Stage 2: Jdirs[j*450 + k*150 + l] = sum_v Jreg[j,v]*shapedirs[v,k,l]
//          Jt[j*3+k]               = sum_v Jreg[j,v]*v_template[v,k]
// split-K over 16 vertex chunks, then reduce.
// ---------------------------------------------------------------
__global__ void jdirs_part(const float* __restrict__ Jreg, const float* __restrict__ sd,
                           const float* __restrict__ vtemp, float* __restrict__ Jpart) {
    const int t = threadIdx.x;
    const int cb = blockIdx.x;
    const int v0 = cb * 314;
    const int v1 = (v0 + 314 < NV) ? (v0 + 314) : NV;
    float* outp = Jpart + cb * 2265;
    if (t < 450) {
        float acc[NJ] = {0.f, 0.f, 0.f, 0.f, 0.f};
        for (int v = v0; v < v1; ++v) {
            float s = sd[v * 450 + t];
#pragma unroll
            for (int j = 0; j < NJ; ++j) acc[j] += Jreg[j * NV + v] * s;
        }
#pragma unroll
        for (int j = 0; j < NJ; ++j) outp[j * 450 + t] = acc[j];
    } else if (t < 465) {
        int idx = t - 450, j = idx / 3, k = idx % 3;
        float acc = 0.f;
        for (int v = v0; v < v1; ++v) acc += Jreg[j * NV + v] * vtemp[v * 3 + k];
        outp[2250 + idx] = acc;
    }
}

__global__ void jdirs_reduce(const float* __restrict__ Jpart,
                             float* __restrict__ Jdirs, float* __restrict__ Jt) {
    int t = blockIdx.x * blockDim.x + threadIdx.x;
    if (t >= 2265) return;
    float acc = 0.f;
#pragma unroll
    for (int c = 0; c < 16; ++c) acc += Jpart[c * 2265 + t];
    if (t < 2250) Jdirs[t] = acc;
    else          Jt[t - 2250] = acc;
}

// ---------------------------------------------------------------
// Stage 3: per-batch small stuff.  Writes:
//   Afull[b][0..149]=betas, [150..151]=0, [152..187]=pose_feature, [188..191]=0
//   Amat [b][5][12] = skinning transforms (rows 0..2 of 4x4)
// ---------------------------------------------------------------
__device__ __forceinline__ void rodrigues1(float p0, float p1, float p2, float* R) {
    float e0 = p0 + 1e-8f, e1 = p1 + 1e-8f, e2 = p2 + 1e-8f;
    float angle = sqrtf(e0 * e0 + e1 * e1 + e2 * e2);
    float inv = 1.0f / angle;
    float rx = p0 * inv, ry = p1 * inv, rz = p2 * inv;
    float s = sinf(angle), c = cosf(angle), oc = 1.0f - c;
    R[0] = 1.0f - oc * (ry * ry + rz * rz);
    R[1] = -s * rz + oc * rx * ry;
    R[2] =  s * ry + oc * rx * rz;
    R[3] =  s * rz + oc * rx * ry;
    R[4] = 1.0f - oc * (rx * rx + rz * rz);
    R[5] = -s * rx + oc * ry * rz;
    R[6] = -s * ry + oc * rx * rz;
    R[7] =  s * rx + oc * ry * rz;
    R[8] = 1.0f - oc * (rx * rx + ry * ry);
}

// O(3x4) = P(3x4) * [Rm(3x3)|t ; 0 0 0 1]
__device__ __forceinline__ void compose34(const float* P, const float* Rm,
                                          const float* t, float* O) {
#pragma unroll
    for (int r = 0; r < 3; ++r) {
#pragma unroll
        for (int c = 0; c < 3; ++c)
            O[r * 4 + c] = P[r * 4 + 0] * Rm[0 * 3 + c] + P[r * 4 + 1] * Rm[1 * 3 + c] +
                           P[r * 4 + 2] * Rm[2 * 3 + c];
        O[r * 4 + 3] = P[r * 4 + 0] * t[0] + P[r * 4 + 1] * t[1] + P[r * 4 + 2] * t[2] +
                       P[r * 4 + 3];
    }
}

__global__ void pose_kernel(const float* __restrict__ shp, const float* __restrict__ expr,
                            const float* __restrict__ neck, const float* __restrict__ jaw,
                            const float* __restrict__ eye,  const float* __restrict__ Jdirs,
                            const float* __restrict__ Jt,   float* __restrict__ Afull,
                            float* __restrict__ Amat) {
    int b = blockIdx.x * blockDim.x + threadIdx.x;
    if (b >= NB) return;
    float* Arow = Afull + (size_t)b * KDIM;
    for (int l = 0; l < NSH; ++l)  Arow[l]       = shp[b * NSH + l];
    for (int l = 0; l < NEXP; ++l) Arow[NSH + l] = expr[b * NEXP + l];
    Arow[150] = 0.f; Arow[151] = 0.f;

    // joints J[j][k] = Jt + Jdirs . betas
    float jacc[15];
#pragma unroll
    for (int m = 0; m < 15; ++m) jacc[m] = Jt[m];
    for (int l = 0; l < 150; ++l) {
        float bl = Arow[l];
#pragma unroll
        for (int m = 0; m < 15; ++m) jacc[m] += Jdirs[m * 150 + l] * bl;
    }
    float J[NJ][3];
#pragma unroll
    for (int j = 0; j < NJ; ++j)
#pragma unroll
        for (int k = 0; k < 3; ++k) J[j][k] = jacc[j * 3 + k];

    // rotations (joint0 pose = 0, matching reference's rodrigues(0))
    float R[NJ][9];
    rodrigues1(0.f, 0.f, 0.f, R[0]);
    rodrigues1(neck[b * 3 + 0], neck[b * 3 + 1], neck[b * 3 + 2], R[1]);
    rodrigues1(jaw[b * 3 + 0],  jaw[b * 3 + 1],  jaw[b * 3 + 2],  R[2]);
    rodrigues1(eye[b * 6 + 0],  eye[b * 6 + 1],  eye[b * 6 + 2],  R[3]);
    rodrigues1(eye[b * 6 + 3],  eye[b * 6 + 4],  eye[b * 6 + 5],  R[4]);

    // pose feature -> A row cols 152..187 ; 188..191 = 0
#pragma unroll
    for (int j = 1; j < NJ; ++j)
#pragma unroll
        for (int i = 0; i < 9; ++i)
            Arow[152 + (j - 1) * 9 + i] = R[j][i] - ((i % 4 == 0) ? 1.0f : 0.0f);
    Arow[188] = 0.f; Arow[189] = 0.f; Arow[190] = 0.f; Arow[191] = 0.f;

    // relative joints (parents: -1,0,1,1,1)
    float rel[NJ][3];
#pragma unroll
    for (int k = 0; k < 3; ++k) {
        rel[0][k] = J[0][k];
        rel[1][k] = J[1][k] - J[0][k];
        rel[2][k] = J[2][k] - J[1][k];
        rel[3][k] = J[3][k] - J[1][k];
        rel[4][k] = J[4][k] - J[1][k];
    }

    // kinematic chain as 3x4 transforms
    float G[NJ][12];
#pragma unroll
    for (int r = 0; r < 3; ++r) {
#pragma unroll
        for (int c = 0; c < 3; ++c) G[0][r * 4 + c] = R[0][r * 3 + c];
        G[0][r * 4 + 3] = rel[0][r];
    }
    compose34(G[0], R[1], rel[1], G[1]);
    compose34(G[1], R[2], rel[2], G[2]);
    compose34(G[1], R[3], rel[3], G[3]);
    compose34(G[1], R[4], rel[4], G[4]);

    // remove rest-pose joint locations, store [5][3x4]
    float* Ab = Amat + (size_t)b * 60;
#pragma unroll
    for (int j = 0; j < NJ; ++j)
#pragma unroll
        for (int r = 0; r < 3; ++r) {
            float tc = G[j][r * 4 + 3] -
                       (G[j][r * 4 + 0] * J[j][0] + G[j][r * 4 + 1] * J[j][1] +
                        G[j][r * 4 + 2] * J[j][2]);
            Ab[j * 12 + r * 4 + 0] = G[j][r * 4 + 0];
            Ab[j * 12 + r * 4 + 1] = G[j][r * 4 + 1];
            Ab[j * 12 + r * 4 + 2] = G[j][r * 4 + 2];
            Ab[j * 12 + r * 4 + 3] = tc;
        }
}

// ---------------------------------------------------------------
// Stage 4: fused WMMA GEMM (f32 16x16x4) + LBS skinning.
// One wave: 32 batches x 48 cols (16 vertices), 6 accumulators.
// 4 waves/block share the SAME vertex tile (same B rows -> L0 reuse),
// covering 128 consecutive batches.  Grid is exact: EXEC all-1s at WMMA.
// Fragment layouts per ISA tables:
//   A(16x4 f32): vgpr0=K0/K2 split by lane-half, vgpr1=K1/K3
//     -> contiguous float2 at row*KDIM + k + 2*(lane>>4); B mirrored.
//   C: elem i of v8f = row M=i+8*(lane>>4), col N=lane&15.
// ---------------------------------------------------------------
__global__ __launch_bounds__(128) void flame_wmma(
    const float* __restrict__ Afull, const float* __restrict__ Bfull,
    const float* __restrict__ Amat,  const float* __restrict__ vtemp,
    const float* __restrict__ lbsw,  float* __restrict__ verts) {
    __shared__ float sAmat[128 * 60];      // 30.0 KB: skinning mats, 128 batches
    __shared__ float sTile[4][32 * 48];    // 24.0 KB: per-wave transpose tiles

    const int lane = threadIdx.x & 31;
    const int wid  = threadIdx.x >> 5;
    const int vt   = blockIdx.x;           // vertex tile 0..313 (exact)
    const int b0   = blockIdx.y * 128;     // block batch base
    const int bw   = b0 + wid * 32;        // wave batch base

    // cooperative load of skinning matrices for this block's 128 batches
    for (int i = threadIdx.x; i < 128 * 60; i += 128) sAmat[i] = Amat[(size_t)b0 * 60 + i];
    __syncthreads();

    const int l16   = lane & 15;
    const int halft = lane >> 4;
    const int c0    = vt * 48;

    const float* aptr0 = Afull + (size_t)(bw + l16) * KDIM + 2 * halft;
    const float* aptr1 = aptr0 + (size_t)16 * KDIM;
    int cA = c0 + l16, cB = c0 + 16 + l16, cC = c0 + 32 + l16;
    int sA = (cA < NC) ? cA : NC - 1;
    int sB = (cB < NC) ? cB : NC - 1;
    int sC = (cC < NC) ? cC : NC - 1;
    const float* bp0 = Bfull + (size_t)sA * KDIM + 2 * halft;
    const float* bp1 = Bfull + (size_t)sB * KDIM + 2 * halft;
    const float* bp2 = Bfull + (size_t)sC * KDIM + 2 * halft;

    v8f z = {0.f,0.f,0.f,0.f,0.f,0.f,0.f,0.f};
    v8f acc00 = z, acc01 = z, acc02 = z;   // batch rows 0..15
    v8f acc10 = z, acc11 = z, acc12 = z;   // batch rows 16..31
#pragma unroll 4
    for (int k = 0; k < KDIM; k += 4) {
        v2f a0 = *(const v2f*)(aptr0 + k);
        v2f a1 = *(const v2f*)(aptr1 + k);
        v2f f0 = *(const v2f*)(bp0 + k);
        v2f f1 = *(const v2f*)(bp1 + k);
        v2f f2 = *(const v2f*)(bp2 + k);
        acc00 = __builtin_amdgcn_wmma_f32_16x16x4_f32(false, a0, false, f0, (short)0, acc00, false, false);
        acc01 = __builtin_amdgcn_wmma_f32_16x16x4_f32(false, a0, false, f1, (short)0, acc01, false, false);
        acc02 = __builtin_amdgcn_wmma_f32_16x16x4_f32(false, a0, false, f2, (short)0, acc02, false, false);
        acc10 = __builtin_amdgcn_wmma_f32_16x16x4_f32(false, a1, false, f0, (short)0, acc10, false, false);
        acc11 = __builtin_amdgcn_wmma_f32_16x16x4_f32(false, a1, false, f1, (short)0, acc11, false, false);
        acc12 = __builtin_amdgcn_wmma_f32_16x16x4_f32(false, a1, false, f2, (short)0, acc12, false, false);
    }

    // store tiles to per-wave LDS (+ v_template), transposing WMMA layout
    float* tile = &sTile[wid][0];
    {
        float vt0 = (cA < NC) ? vtemp[cA] : 0.0f;
        float vt1 = (cB < NC) ? vtemp[cB] : 0.0f;
        float vt2 = (cC < NC) ? vtemp[cC] : 0.0f;
#pragma unroll
        for (int i = 0; i < 8; ++i) {
            int m0 = i + 8 * halft;          // rows 0..15
            int m1 = m0 + 16;                // rows 16..31
            tile[m0 * 48 +  0 + l16] = acc00[i] + vt0;
            tile[m0 * 48 + 16 + l16] = acc01[i] + vt1;
            tile[m0 * 48 + 32 + l16] = acc02[i] + vt2;
            tile[m1 * 48 +  0 + l16] = acc10[i] + vt0;
            tile[m1 * 48 + 16 + l16] = acc11[i] + vt1;
            tile[m1 * 48 + 32 + l16] = acc12[i] + vt2;
        }
    }
    // same-wave LDS executes in-order; wait + compiler barrier before re-read
    asm volatile("s_wait_dscnt 0" ::: "memory");

    // LBS skinning: 32 batches x 16 vertices = 512 instances -> 16 per lane
#pragma unroll
    for (int it = 0; it < 16; ++it) {
        int inst = it * 32 + lane;
        int bl = inst >> 4;       // local batch row 0..31 (within wave)
        int vl = inst & 15;       // local vertex  0..15
        int v  = vt * 16 + vl;
        if (v < NV) {
            float x = tile[bl * 48 + vl * 3 + 0];
            float y = tile[bl * 48 + vl * 3 + 1];
            float z2 = tile[bl * 48 + vl * 3 + 2];
            const float* wrow = lbsw + (size_t)v * NJ;
            float T[12];
#pragma unroll
            for (int i = 0; i < 12; ++i) T[i] = 0.f;
#pragma unroll
            for (int j = 0; j < NJ; ++j) {
                float w = wrow[j];
                const float* Aj = &sAmat[(wid * 32 + bl) * 60 + j * 12];
#pragma unroll
                for (int i = 0; i < 12; ++i) T[i] += w * Aj[i];
            }
            float ox = T[0] * x + T[1] * y + T[2]  * z2 + T[3];
            float oy = T[4] * x + T[5] * y + T[6]  * z2 + T[7];
            float oz = T[8] * x + T[9] * y + T[10] * z2 + T[11];
            float* o = verts + (size_t)(bw + bl) * NC + (size_t)v * 3;
            o[0] = ox; o[1] = oy; o[2] = oz;
        }
    }
}

// ---------------------------------------------------------------
extern "C" void kernel_launch(void* const* d_in, const int* in_sizes, int n_in,
                              void* d_out, int out_size, void* d_ws, size_t ws_size,
                              hipStream_t stream) {
    const float* shp   = (const float*)d_in[0];
    const float* expr  = (const float*)d_in[1];
    const float* neck  = (const float*)d_in[2];
    const float* jaw   = (const float*)d_in[3];
    const float* eye   = (const float*)d_in[4];
    const float* vtemp = (const float*)d_in[5];
    const float* sdirs = (const float*)d_in[6];
    const float* pdirs = (const float*)d_in[7];
    const float* Jreg  = (const float*)d_in[8];
    const float* lbsw  = (const float*)d_in[9];
    // d_in[10] = parents (static 5-joint tree, baked in)

    float* ws    = (float*)d_ws;
    float* Afull = ws;                                   // NB*KDIM      = 393216
    float* Bfull = Afull + (size_t)NB * KDIM;            // NC*KDIM      = 2893248
    float* Amat  = Bfull + (size_t)NC * KDIM;            // NB*60        = 122880
    float* Jdirs = Amat  + (size_t)NB * 60;              // 2250
    float* Jt    = Jdirs + 2250;                         // 15
    float* Jpart = Jt    + 15;                           // 16*2265      = 36240
    float* verts = (float*)d_out;

    build_B<<<2048, 256, 0, stream>>>(sdirs, pdirs, Bfull);
    jdirs_part<<<16, 512, 0, stream>>>(Jreg, sdirs, vtemp, Jpart);
    jdirs_reduce<<<9, 256, 0, stream>>>(Jpart, Jdirs, Jt);
    pose_kernel<<<NB / 64, 64, 0, stream>>>(shp, expr, neck, jaw, eye, Jdirs, Jt, Afull, Amat);
    dim3 grid(VTILES, NB / 128);                         // 314 x 16, exact
    flame_wmma<<<grid, 128, 0, stream>>>(Afull, Bfull, Amat, vtemp, lbsw, verts);
}